// TestAllReduceRMSNormModel_69784628625430
// MI455X (gfx1250) — compile-verified
//
#include <hip/hip_runtime.h>
#include <stdint.h>

typedef float v4f __attribute__((ext_vector_type(4)));

#define BLOCK 256
#define EPS   1e-6f

// NF4: float4 chunks per thread (hidden = NF4 * BLOCK * 4), compile-time.
// TP:  tensor-parallel ranks, compile-time (0 = use runtime tp argument).
template<int NF4, int TP>
__global__ __launch_bounds__(BLOCK)
void fused_relu_allreduce_rmsnorm(const float* __restrict__ x,
                                  const float* __restrict__ w,
                                  float* __restrict__ out,
                                  int tp_rt, int tokens, int hidden)
{
    __shared__ float s_w[NF4 * BLOCK * 4];     // async-staged weight
    __shared__ float s_part[BLOCK / 32];       // 8 wave partials (wave32)
    __shared__ float s_scale;

    const int    tid    = threadIdx.x;
    const size_t rowOff = (size_t)blockIdx.x * (size_t)hidden;
    const int    tp     = TP ? TP : tp_rt;

    // ---- 1) Async weight -> LDS (ASYNCcnt), overlapped with x streaming.
    //      Each thread fetches exactly the 16B it consumes later, so the
    //      per-wave s_wait_asynccnt 0 is the only synchronization needed.
#pragma unroll
    for (int i = 0; i < NF4; ++i) {
        const int h4  = i * BLOCK + tid;                     // float4 index
        uint32_t  lds = (uint32_t)(size_t)&s_w[h4 * 4];      // low 32b = LDS addr
        uint64_t  ga  = (uint64_t)(const void*)(w + (size_t)h4 * 4);
        asm volatile("global_load_async_to_lds_b128 %0, %1, off"
                     :: "v"(lds), "v"(ga) : "memory");
    }

    // ---- 2) r = sum over ranks of relu(x); kept entirely in registers.
    //      NT loads: 512 MB streamed once, keep it out of L2's working set.
    v4f acc[NF4];
#pragma unroll
    for (int i = 0; i < NF4; ++i) acc[i] = (v4f)0.0f;

#pragma unroll
    for (int rk = 0; rk < (TP ? TP : 8); ++rk) {
        if (TP || rk < tp) {
            const v4f* xp = (const v4f*)(x + (size_t)rk * (size_t)tokens * (size_t)hidden
                                           + rowOff);
#pragma unroll
            for (int i = 0; i < NF4; ++i) {
                v4f v = __builtin_nontemporal_load(&xp[i * BLOCK + tid]);
                v4f z;
                z.x = v.x > 0.f ? v.x : 0.f;
                z.y = v.y > 0.f ? v.y : 0.f;
                z.z = v.z > 0.f ? v.z : 0.f;
                z.w = v.w > 0.f ? v.w : 0.f;
                acc[i] += z;
            }
        }
    }

    // ---- 3) sum of squares: thread -> wave32 shuffle tree -> cross-wave LDS
    float ssq = 0.f;
#pragma unroll
    for (int i = 0; i < NF4; ++i)
        ssq += acc[i].x * acc[i].x + acc[i].y * acc[i].y
             + acc[i].z * acc[i].z + acc[i].w * acc[i].w;

#pragma unroll
    for (int m = 16; m >= 1; m >>= 1)          // wave32: masks 16..1 only
        ssq += __shfl_xor(ssq, m, 32);

    if ((tid & 31) == 0) s_part[tid >> 5] = ssq;
    __syncthreads();
    if (tid == 0) {
        float t = 0.f;
#pragma unroll
        for (int i = 0; i < BLOCK / 32; ++i) t += s_part[i];
        float v = t / (float)hidden + EPS;
        float r = rsqrtf(v);
        r = r * (1.5f - 0.5f * v * r * r);     // one Newton step for accuracy
        s_scale = r;
    }
    __syncthreads();
    const float scale = s_scale;

    // ---- 4) wait for our async weight tiles, scale, NT-store the output.
    asm volatile("s_wait_asynccnt 0" ::: "memory");
    const v4f* wp = (const v4f*)s_w;
    v4f*       op = (v4f*)(out + rowOff);
#pragma unroll
    for (int i = 0; i < NF4; ++i) {
        const int h4 = i * BLOCK + tid;
        v4f ww = wp[h4];                        // ds_load_b128 from staged weight
        v4f o  = acc[i] * scale * ww;
        __builtin_nontemporal_store(o, &op[h4]);
    }
}

template<int NF4>
static void launch_nf4(const float* x, const float* w, float* out,
                       int tp, int tokens, int hidden, hipStream_t stream) {
    dim3 grid(tokens), block(BLOCK);
    if (tp == 4) {
        hipLaunchKernelGGL((fused_relu_allreduce_rmsnorm<NF4, 4>),
                           grid, block, 0, stream, x, w, out, tp, tokens, hidden);
    } else {
        hipLaunchKernelGGL((fused_relu_allreduce_rmsnorm<NF4, 0>),
                           grid, block, 0, stream, x, w, out, tp, tokens, hidden);
    }
}

extern "C" void kernel_launch(void* const* d_in, const int* in_sizes, int n_in,
                              void* d_out, int out_size, void* d_ws, size_t ws_size,
                              hipStream_t stream) {
    const float* x   = (const float*)d_in[0];   // [tp, tokens, hidden] f32
    const float* w   = (const float*)d_in[1];   // [hidden] f32
    float*       out = (float*)d_out;           // [tokens, hidden] f32

    const int hidden = in_sizes[1];             // 4096
    const int tokens = out_size / hidden;       // 8192
    const int tp     = in_sizes[0] / out_size;  // 4
    const int nf4    = hidden / (BLOCK * 4);    // 4 for hidden=4096

    switch (nf4) {
        case 1: launch_nf4<1>(x, w, out, tp, tokens, hidden, stream); break;
        case 2: launch_nf4<2>(x, w, out, tp, tokens, hidden, stream); break;
        case 3: launch_nf4<3>(x, w, out, tp, tokens, hidden, stream); break;
        case 4: launch_nf4<4>(x, w, out, tp, tokens, hidden, stream); break;
        case 5: launch_nf4<5>(x, w, out, tp, tokens, hidden, stream); break;
        case 6: launch_nf4<6>(x, w, out, tp, tokens, hidden, stream); break;
        case 7: launch_nf4<7>(x, w, out, tp, tokens, hidden, stream); break;
        default: launch_nf4<8>(x, w, out, tp, tokens, hidden, stream); break;
    }
}